// Sampler_54065048323066
// MI455X (gfx1250) — compile-verified
//
#include <hip/hip_runtime.h>
#include <hip/hip_bf16.h>

#define LN2F 0.6931471805599453f
#define BLOCK_T 512
#define NWAVES (BLOCK_T / 32)
#define PIPE 4                      // async pipeline depth (tiles in flight)

#if defined(__has_builtin)
#if __has_builtin(__builtin_amdgcn_global_load_async_to_lds_b128) && \
    __has_builtin(__builtin_amdgcn_s_wait_asynccnt)
#define SAMPLER_ASYNC 1
#endif
#endif
#ifndef SAMPLER_ASYNC
#define SAMPLER_ASYNC 0
#endif

#if SAMPLER_ASYNC
// Pointee type required by the builtin (per hipcc diagnostic):
//   int __attribute__((vector_size(16)))
typedef int v4i_raw __attribute__((vector_size(16)));
typedef __attribute__((address_space(1))) v4i_raw* gv4i_ptr;  // global
typedef __attribute__((address_space(3))) v4i_raw* lv4i_ptr;  // LDS
#endif

__device__ __forceinline__ unsigned rotl32(unsigned x, unsigned n) {
  return (x << n) | (x >> (32u - n));
}

// JAX threefry2x32 with key = jax.random.key(42) -> (0, 42), 20 rounds.
__device__ __forceinline__ void threefry2x32_42(unsigned c0, unsigned c1,
                                                unsigned& o0, unsigned& o1) {
  const unsigned ks0 = 0u, ks1 = 42u;
  const unsigned ks2 = 0x1BD11BDAu ^ ks0 ^ ks1;
  unsigned x0 = c0 + ks0;
  unsigned x1 = c1 + ks1;
#define TF_ROUND(r) { x0 += x1; x1 = rotl32(x1, (r)); x1 ^= x0; }
  TF_ROUND(13u) TF_ROUND(15u) TF_ROUND(26u) TF_ROUND(6u)
  x0 += ks1; x1 += ks2 + 1u;
  TF_ROUND(17u) TF_ROUND(29u) TF_ROUND(16u) TF_ROUND(24u)
  x0 += ks2; x1 += ks0 + 2u;
  TF_ROUND(13u) TF_ROUND(15u) TF_ROUND(26u) TF_ROUND(6u)
  x0 += ks0; x1 += ks1 + 3u;
  TF_ROUND(17u) TF_ROUND(29u) TF_ROUND(16u) TF_ROUND(24u)
  x0 += ks1; x1 += ks2 + 4u;
  TF_ROUND(13u) TF_ROUND(15u) TF_ROUND(26u) TF_ROUND(6u)
  x0 += ks2; x1 += ks0 + 5u;
#undef TF_ROUND
  o0 = x0; o1 = x1;
}

// score = logit/T - log(max(Exp1(bits), 1e-10)); Exp1 = -log1p(-u),
// u = bitcast(0x3f800000 | bits>>9) - 1  (JAX uniform bit trick).
// 1-u == 2-f exactly, so Exp1 = -ln2 * log2(2 - f).
__device__ __forceinline__ float gumbel_score(float logit, float invT, unsigned bits) {
  float f = __uint_as_float(0x3f800000u | (bits >> 9));
  float one_minus_u = 2.0f - f;                 // (0, 1]
  float E = -LN2F * __log2f(one_minus_u);       // Exp(1) sample, >= 0
  E = fmaxf(E, 1e-10f);
  return __builtin_fmaf(logit, invT, -LN2F * __log2f(E));
}

__global__ __launch_bounds__(BLOCK_T) void sampler_argmax_kernel(
    const float* __restrict__ logits, const float* __restrict__ temps,
    int* __restrict__ out, int V, int NV4, int B) {
  const int tid = threadIdx.x;
  const int halfB = B >> 1;
  const int row0 = blockIdx.x;          // 0..halfB-1
  const int row1 = row0 + halfB;        // paired row shares threefry blocks
  const unsigned halfN = (unsigned)halfB * (unsigned)V;

  const float invT0 = 1.0f / temps[row0];
  const float invT1 = 1.0f / temps[row1];
  const float4* __restrict__ p0 = (const float4*)(logits + (size_t)row0 * (size_t)V);
  const float4* __restrict__ p1 = (const float4*)(logits + (size_t)row1 * (size_t)V);

  float best0 = -INFINITY, best1 = -INFINITY;
  int bi0 = 0, bi1 = 0;

  const unsigned rowBase0 = (unsigned)row0 * (unsigned)V;

  auto consume = [&](float4 a, float4 b, int vec) {
    const int colBase = vec << 2;
    const unsigned i0base = rowBase0 + (unsigned)colBase;
    const float* av = (const float*)&a;
    const float* bv = (const float*)&b;
#pragma unroll
    for (int j = 0; j < 4; ++j) {
      unsigned o0, o1;
      threefry2x32_42(i0base + (unsigned)j, i0base + (unsigned)j + halfN, o0, o1);
      float s0 = gumbel_score(av[j], invT0, o0);
      float s1 = gumbel_score(bv[j], invT1, o1);
      int col = colBase + j;
      if (s0 > best0) { best0 = s0; bi0 = col; }
      if (s1 > best1) { best1 = s1; bi1 = col; }
    }
  };

  const int count = (NV4 + BLOCK_T - 1) / BLOCK_T;

#if SAMPLER_ASYNC
  // PIPE-deep async global->LDS streaming (CDNA5 ASYNCcnt path).
  // Each lane async-loads its own 16B slot, so the wave only waits on its
  // own ASYNCcnt -- no cross-wave barriers in the hot loop. PIPE=4 keeps
  // 8 async b128 loads (4 KB/wave) in flight to cover HBM latency.
  __shared__ __align__(16) float4 tA[PIPE][BLOCK_T];
  __shared__ __align__(16) float4 tB[PIPE][BLOCK_T];

  auto issueTile = [&](int k) {
    int vec = k * BLOCK_T + tid;
    if (vec >= NV4) vec = NV4 - 1;  // clamped address; consumption is guarded
    const int par = k & (PIPE - 1);
    __builtin_amdgcn_global_load_async_to_lds_b128(
        (gv4i_ptr)(p0 + vec), (lv4i_ptr)&tA[par][tid], 0, 0);
    __builtin_amdgcn_global_load_async_to_lds_b128(
        (gv4i_ptr)(p1 + vec), (lv4i_ptr)&tB[par][tid], 0, 0);
  };

#pragma unroll
  for (int i = 0; i < PIPE - 1; ++i) {
    if (i < count) issueTile(i);
  }
  for (int k = 0; k < count; ++k) {
    if (k + PIPE - 1 < count) {
      issueTile(k + PIPE - 1);
      // 2*(PIPE-1) younger loads may remain; the 2 oldest (tile k) are done.
      __builtin_amdgcn_s_wait_asynccnt(2 * (PIPE - 1));
    } else {
      __builtin_amdgcn_s_wait_asynccnt(0);  // tail: drain everything
    }
    asm volatile("" ::: "memory");  // keep LDS reads after the wait
    const int par = k & (PIPE - 1);
    float4 a = tA[par][tid];
    float4 b = tB[par][tid];
    int vec = k * BLOCK_T + tid;
    if (vec < NV4) consume(a, b, vec);
  }
#else
  for (int k = 0; k < count; ++k) {
    int vec = k * BLOCK_T + tid;
    if (vec + 2 * BLOCK_T < NV4) {
      __builtin_prefetch(p0 + vec + 2 * BLOCK_T, 0, 0);
      __builtin_prefetch(p1 + vec + 2 * BLOCK_T, 0, 0);
    }
    if (vec < NV4) {
      float4 a = p0[vec];
      float4 b = p1[vec];
      consume(a, b, vec);
    }
  }
#endif

  // wave32 argmax reduction (tie -> smaller index, matching argmax-first)
#pragma unroll
  for (int off = 16; off > 0; off >>= 1) {
    float s0 = __shfl_xor(best0, off, 32);
    int   j0 = __shfl_xor(bi0,   off, 32);
    if (s0 > best0 || (s0 == best0 && j0 < bi0)) { best0 = s0; bi0 = j0; }
    float s1 = __shfl_xor(best1, off, 32);
    int   j1 = __shfl_xor(bi1,   off, 32);
    if (s1 > best1 || (s1 == best1 && j1 < bi1)) { best1 = s1; bi1 = j1; }
  }

  __shared__ float sb0[NWAVES], sb1[NWAVES];
  __shared__ int   si0[NWAVES], si1[NWAVES];
  const int wave = tid >> 5, lane = tid & 31;
  if (lane == 0) {
    sb0[wave] = best0; si0[wave] = bi0;
    sb1[wave] = best1; si1[wave] = bi1;
  }
  __syncthreads();
  if (tid == 0) {
    float b0 = sb0[0]; int j0 = si0[0];
    float b1 = sb1[0]; int j1 = si1[0];
#pragma unroll
    for (int w = 1; w < NWAVES; ++w) {
      if (sb0[w] > b0 || (sb0[w] == b0 && si0[w] < j0)) { b0 = sb0[w]; j0 = si0[w]; }
      if (sb1[w] > b1 || (sb1[w] == b1 && si1[w] < j1)) { b1 = sb1[w]; j1 = si1[w]; }
    }
    out[row0] = j0;
    out[row1] = j1;
  }
}

extern "C" void kernel_launch(void* const* d_in, const int* in_sizes, int n_in,
                              void* d_out, int out_size, void* d_ws, size_t ws_size,
                              hipStream_t stream) {
  (void)n_in; (void)d_ws; (void)ws_size; (void)out_size;
  const float* logits = (const float*)d_in[0];   // (B, V) fp32
  const float* temps  = (const float*)d_in[1];   // (B,)  fp32
  int* out = (int*)d_out;                        // (B,) int32 argmax indices

  const int BV = in_sizes[0];
  const int B  = in_sizes[1];
  const int V  = BV / B;
  const int NV4 = V >> 2;                        // float4 vectors per row

  dim3 grid(B / 2);        // one block per (r, r+B/2) row pair
  dim3 block(BLOCK_T);     // 16 wave32 waves
  sampler_argmax_kernel<<<grid, block, 0, stream>>>(logits, temps, out, V, NV4, B);
}